// EnhancedIPA_27118423507358
// MI455X (gfx1250) — compile-verified
//
#include <hip/hip_runtime.h>
#include <math.h>

#define LSEQ 320
#define CSD  384
#define CZD  128
#define CHD  16
#define NH   12
#define PQN  4
#define PVN  8
#define EPSF 1e-8f

typedef __attribute__((ext_vector_type(2))) float v2f;
typedef __attribute__((ext_vector_type(8))) float v8f;

// ---------------------------------------------------------------------------
// Generic fp32 WMMA GEMM + bias:  C[M x N] = A[M x K] @ B[K x N] + bias[N]
// one wave computes one 16x16 tile using V_WMMA_F32_16X16X4_F32, K%4==0.
// ---------------------------------------------------------------------------
__global__ __launch_bounds__(32) void wmma_gemm_bias(
    const float* __restrict__ A, const float* __restrict__ B,
    const float* __restrict__ bias, float* __restrict__ C,
    int K, int lda, int ldb, int ldc)
{
    const int lane = threadIdx.x;
    const int row  = lane & 15;   // M row (A) / N col (B,C)
    const int hi   = lane >> 4;   // K sub-slot select
    const int m0   = blockIdx.y * 16;
    const int n0   = blockIdx.x * 16;

    const float* Arow = A + (size_t)(m0 + row) * lda + 2 * hi;
    const float* Bp   = B + (size_t)(2 * hi) * ldb + n0 + row;

    v8f acc = {};
    for (int k0 = 0; k0 < K; k0 += 4) {
        v2f a = *(const v2f*)(Arow + k0);            // A[m][k0+2hi], A[m][k0+2hi+1]
        v2f b;
        b.x = Bp[(size_t)k0 * ldb];                  // B[k0+2hi][n]
        b.y = Bp[(size_t)(k0 + 1) * ldb];            // B[k0+2hi+1][n]
        acc = __builtin_amdgcn_wmma_f32_16x16x4_f32(
            false, a, false, b, (short)0, acc, false, false);
    }

    const float bv = bias[n0 + row];
    float* Cp = C + (size_t)(m0 + hi * 8) * ldc + n0 + row;
#pragma unroll
    for (int r = 0; r < 8; ++r) Cp[(size_t)r * ldc] = acc[r] + bv;
}

// ---------------------------------------------------------------------------
// Strided-batched fp32 WMMA GEMM (no bias), row-guarded in M (for M=12 pads).
// C[b][m][n] = sum_k A[b][m][k] * B[b][k][n]
// ---------------------------------------------------------------------------
__global__ __launch_bounds__(32) void wmma_gemm_batched(
    const float* __restrict__ A, long long a_row, long long a_batch,
    const float* __restrict__ B, long long b_row, long long b_batch,
    float* __restrict__ C, long long c_row, long long c_batch,
    int M, int K)
{
    const int lane = threadIdx.x;
    const int row  = lane & 15;
    const int hi   = lane >> 4;
    const int m0   = blockIdx.y * 16;
    const int n0   = blockIdx.x * 16;

    const float* Ab = A + (long long)blockIdx.z * a_batch;
    const float* Bb = B + (long long)blockIdx.z * b_batch;
    float*       Cb = C + (long long)blockIdx.z * c_batch;

    const int  m   = m0 + row;
    const bool mok = (m < M);
    const float* Arow = Ab + (long long)m * a_row + 2 * hi;
    const float* Bp   = Bb + (long long)(2 * hi) * b_row + n0 + row;

    v8f acc = {};
    for (int k0 = 0; k0 < K; k0 += 4) {
        v2f a = {0.f, 0.f};
        if (mok) a = *(const v2f*)(Arow + k0);
        v2f b;
        b.x = Bp[(long long)k0 * b_row];
        b.y = Bp[(long long)(k0 + 1) * b_row];
        if (((k0 & 31) == 0) && (k0 + 32) < K)       // stream-ahead hint for z
            __builtin_prefetch(Bp + (long long)(k0 + 32) * b_row, 0, 1);
        acc = __builtin_amdgcn_wmma_f32_16x16x4_f32(
            false, a, false, b, (short)0, acc, false, false);
    }
#pragma unroll
    for (int r = 0; r < 8; ++r) {
        int mr = m0 + hi * 8 + r;
        if (mr < M) Cb[(long long)mr * c_row + n0 + row] = acc[r];
    }
}

// ---------------------------------------------------------------------------
// proc_pts + gate: rotate/translate q/k/v points into global frame.
// ---------------------------------------------------------------------------
__global__ void pts_gate_kernel(
    const float* __restrict__ pqp, const float* __restrict__ pkvp,
    const float* __restrict__ pg, const float* __restrict__ rot,
    const float* __restrict__ trans,
    float* __restrict__ q_pts, float* __restrict__ k_pts, float* __restrict__ v_pts)
{
    const int nq = LSEQ * NH * PQN;                 // 15360
    const int nv = LSEQ * NH * PVN;                 // 30720
    int idx = blockIdx.x * blockDim.x + threadIdx.x;
    if (idx >= 2 * nq + nv) return;

    const float* src; float* dst; int l; float g = 1.f;
    if (idx < nq) {                                  // q points (gated)
        l = idx / (NH * PQN); int r = idx % (NH * PQN);
        src = pqp   + (size_t)l * 6 * NH * PQN + r * 6;
        dst = q_pts + (size_t)l * 6 * NH * PQN + r * 6;
        g = 1.f / (1.f + __expf(-pg[(size_t)l * NH * PQN + r]));
    } else if (idx < 2 * nq) {                       // k points
        int i = idx - nq; l = i / (NH * PQN); int r = i % (NH * PQN);
        src = pkvp  + (size_t)l * 6 * NH * (PQN + PVN) + r * 6;
        dst = k_pts + (size_t)l * 6 * NH * PQN + r * 6;
    } else {                                         // v points
        int i = idx - 2 * nq; l = i / (NH * PVN); int r = i % (NH * PVN);
        src = pkvp  + (size_t)l * 6 * NH * (PQN + PVN) + 6 * NH * PQN + r * 6;
        dst = v_pts + (size_t)l * 6 * NH * PVN + r * 6;
    }
    const float* R = rot + (size_t)l * 9;
    const float* t = trans + (size_t)l * 3;
    float p0 = src[0], p1 = src[1], p2 = src[2];
    float d0 = src[3], d1 = src[4], d2 = src[5];
#pragma unroll
    for (int i = 0; i < 3; ++i) {
        float c  = R[i*3+0]*p0 + R[i*3+1]*p1 + R[i*3+2]*p2 + t[i];
        float dd = R[i*3+0]*d0 + R[i*3+1]*d1 + R[i*3+2]*d2;
        dst[i]     = g * c;
        dst[3 + i] = g * dd;
    }
}

// ---------------------------------------------------------------------------
// Attention logits (scalar + pos + dir) + masked softmax over k.
// One block per (q, h); 320 threads (10 wave32s), thread = key index.
// ---------------------------------------------------------------------------
__global__ __launch_bounds__(320) void att_kernel(
    const float* __restrict__ proj_q, const float* __restrict__ proj_kv,
    const float* __restrict__ q_pts, const float* __restrict__ k_pts,
    const float* __restrict__ fmask, const float* __restrict__ geomw,
    const float* __restrict__ dirw, const float* __restrict__ headw,
    float* __restrict__ att)
{
    const int qi = blockIdx.x;
    const int h  = blockIdx.y;
    const int tid = threadIdx.x;       // key index kj

    __shared__ float sq[16];           // q scalar features for (qi,h)
    __shared__ float sqp[24];          // q points (PQ=4 x 6)
    __shared__ float snq[4];           // ||qd_p||
    __shared__ float red[LSEQ];

    if (tid < 16)       sq[tid]       = proj_q[(size_t)qi * (NH*CHD) + h * CHD + tid];
    else if (tid < 40)  sqp[tid - 16] = q_pts[(size_t)qi * (6*NH*PQN) + h * 24 + (tid - 16)];
    __syncthreads();
    if (tid < 4) {
        const float* d = &sqp[tid * 6 + 3];
        snq[tid] = sqrtf(d[0]*d[0] + d[1]*d[1] + d[2]*d[2]);
    }
    __syncthreads();

    const int kj = tid;
    // scalar attention: q . k / sqrt(CH)
    const float* kvec = proj_kv + (size_t)kj * (2*NH*CHD) + h * CHD;
    float sc = 0.f;
#pragma unroll
    for (int c = 0; c < CHD; ++c) sc += sq[c] * kvec[c];
    sc *= 0.25f;                      // 1/sqrt(16)

    float kpv[24];
    const float* kp = k_pts + (size_t)kj * (6*NH*PQN) + h * 24;
#pragma unroll
    for (int i = 0; i < 24; ++i) kpv[i] = kp[i];

    float pos = 0.f, dir = 0.f;
#pragma unroll
    for (int p = 0; p < PQN; ++p) {
        // point distance term
        float dx = sqp[p*6+0] - kpv[p*6+0];
        float dy = sqp[p*6+1] - kpv[p*6+1];
        float dz = sqp[p*6+2] - kpv[p*6+2];
        float d = sqrtf(dx*dx + dy*dy + dz*dz);
        pos += d + __logf(d + EPSF) + 1.f / (d + EPSF);
        // direction term
        const float k0 = kpv[p*6+3], k1 = kpv[p*6+4], k2 = kpv[p*6+5];
        float nk = sqrtf(k0*k0 + k1*k1 + k2*k2);
        float dotp = sqp[p*6+3]*k0 + sqp[p*6+4]*k1 + sqp[p*6+5]*k2;
        float csum = 0.f;
#pragma unroll
        for (int pp = 0; pp < PQN; ++pp) {
            const float q0 = sqp[pp*6+3], q1 = sqp[pp*6+4], q2 = sqp[pp*6+5];
            float cx = q1*k2 - q2*k1;
            float cy = q2*k0 - q0*k2;
            float cz = q0*k1 - q1*k0;
            csum += sqrtf(cx*cx + cy*cy + cz*cz);
        }
        dir += dotp - (csum * 0.25f) / (snq[p] * nk + EPSF);
    }
    pos *= 0.25f;  dir *= 0.25f;      // mean over PQ

    float logit = sc + dirw[0] * (geomw[0] * pos + geomw[1] * dir);
    logit *= headw[h];
    logit += 100000.f * (fmask[qi] * fmask[kj] - 1.f);

    // block softmax over 320 keys
    red[tid] = logit; __syncthreads();
    for (int s = 256; s > 0; s >>= 1) {
        if (tid < s && tid + s < LSEQ) red[tid] = fmaxf(red[tid], red[tid + s]);
        __syncthreads();
    }
    float mx = red[0]; __syncthreads();
    float e = __expf(logit - mx);
    red[tid] = e; __syncthreads();
    for (int s = 256; s > 0; s >>= 1) {
        if (tid < s && tid + s < LSEQ) red[tid] += red[tid + s];
        __syncthreads();
    }
    att[((size_t)h * LSEQ + qi) * LSEQ + kj] = e / red[0];
}

// ---------------------------------------------------------------------------
// Invert frames on o_geom -> [lc(3), ld_norm(3), ||lc||] into concat buffer.
// ---------------------------------------------------------------------------
__global__ void frame_inv_kernel(
    const float* __restrict__ og, const float* __restrict__ rot,
    const float* __restrict__ trans, float* __restrict__ concat)
{
    int idx = blockIdx.x * blockDim.x + threadIdx.x;
    if (idx >= LSEQ * NH * PVN) return;
    int l = idx / (NH * PVN);
    int r = idx % (NH * PVN);
    const float* s6 = og + (size_t)l * (NH*PVN*6) + r * 6;
    const float* R = rot + (size_t)l * 9;
    const float* t = trans + (size_t)l * 3;
    float c0 = s6[0] - t[0], c1 = s6[1] - t[1], c2 = s6[2] - t[2];
    float lc[3], ld[3];
#pragma unroll
    for (int i = 0; i < 3; ++i) {
        lc[i] = R[0*3+i]*c0   + R[1*3+i]*c1   + R[2*3+i]*c2;     // R^T (c - t)
        ld[i] = R[0*3+i]*s6[3] + R[1*3+i]*s6[4] + R[2*3+i]*s6[5]; // R^T dirs
    }
    float nl  = sqrtf(ld[0]*ld[0] + ld[1]*ld[1] + ld[2]*ld[2]);
    float inv = 1.f / fmaxf(nl, 1e-12f);
    float ln  = sqrtf(lc[0]*lc[0] + lc[1]*lc[1] + lc[2]*lc[2]);
    float* dst = concat + (size_t)l * 2400 + NH*CHD + r * 7;
    dst[0] = lc[0]; dst[1] = lc[1]; dst[2] = lc[2];
    dst[3] = ld[0]*inv; dst[4] = ld[1]*inv; dst[5] = ld[2]*inv;
    dst[6] = ln;
}

// ---------------------------------------------------------------------------
extern "C" void kernel_launch(void* const* d_in, const int* in_sizes, int n_in,
                              void* d_out, int out_size, void* d_ws, size_t ws_size,
                              hipStream_t stream) {
    const float* s     = (const float*)d_in[0];
    const float* rot   = (const float*)d_in[1];
    const float* trans = (const float*)d_in[2];
    const float* z     = (const float*)d_in[3];
    const float* fmask = (const float*)d_in[4];
    const float* Wq    = (const float*)d_in[5];
    const float* bq    = (const float*)d_in[6];
    const float* Wkv   = (const float*)d_in[7];
    const float* bkv   = (const float*)d_in[8];
    const float* Wqp   = (const float*)d_in[9];
    const float* bqp   = (const float*)d_in[10];
    const float* Wkvp  = (const float*)d_in[11];
    const float* bkvp  = (const float*)d_in[12];
    const float* Wg    = (const float*)d_in[13];
    const float* bg    = (const float*)d_in[14];
    const float* geomw = (const float*)d_in[15];
    const float* dirw  = (const float*)d_in[16];
    const float* headw = (const float*)d_in[17];
    const float* Wo    = (const float*)d_in[18];
    const float* bo    = (const float*)d_in[19];
    float* out = (float*)d_out;

    float* w = (float*)d_ws;
    float* proj_q   = w; w += LSEQ * 192;
    float* proj_kv  = w; w += LSEQ * 384;
    float* proj_qp  = w; w += LSEQ * 288;
    float* proj_kvp = w; w += LSEQ * 864;
    float* proj_g   = w; w += LSEQ * 48;
    float* q_pts    = w; w += LSEQ * 288;
    float* k_pts    = w; w += LSEQ * 288;
    float* v_pts    = w; w += LSEQ * 576;
    float* o_geom   = w; w += LSEQ * 576;
    float* att      = w; w += (size_t)NH * LSEQ * LSEQ;
    float* concat   = w; w += LSEQ * 2400;

    const dim3 wv(32);
    const int mt = LSEQ / 16;   // 20

    // projections: s @ W + b
    wmma_gemm_bias<<<dim3(192/16, mt), wv, 0, stream>>>(s, Wq,   bq,   proj_q,   CSD, CSD, 192, 192);
    wmma_gemm_bias<<<dim3(384/16, mt), wv, 0, stream>>>(s, Wkv,  bkv,  proj_kv,  CSD, CSD, 384, 384);
    wmma_gemm_bias<<<dim3(288/16, mt), wv, 0, stream>>>(s, Wqp,  bqp,  proj_qp,  CSD, CSD, 288, 288);
    wmma_gemm_bias<<<dim3(864/16, mt), wv, 0, stream>>>(s, Wkvp, bkvp, proj_kvp, CSD, CSD, 864, 864);
    wmma_gemm_bias<<<dim3( 48/16, mt), wv, 0, stream>>>(s, Wg,   bg,   proj_g,   CSD, CSD,  48,  48);

    // frame-transform points + gate
    {
        int n = 2 * LSEQ * NH * PQN + LSEQ * NH * PVN;
        pts_gate_kernel<<<(n + 255) / 256, 256, 0, stream>>>(
            proj_qp, proj_kvp, proj_g, rot, trans, q_pts, k_pts, v_pts);
    }

    // attention logits + softmax
    att_kernel<<<dim3(LSEQ, NH), 320, 0, stream>>>(
        proj_q, proj_kv, q_pts, k_pts, fmask, geomw, dirw, headw, att);

    // o_scalar = att @ v   (batched over heads) -> concat[:, 0:192]
    wmma_gemm_batched<<<dim3(1, mt, NH), wv, 0, stream>>>(
        att, (long long)LSEQ, (long long)LSEQ * LSEQ,
        proj_kv + 192, 384LL, 16LL,
        concat, 2400LL, 16LL,
        LSEQ, LSEQ);

    // o_geom = att @ v_pts (batched over heads) -> o_geom scratch
    wmma_gemm_batched<<<dim3(3, mt, NH), wv, 0, stream>>>(
        att, (long long)LSEQ, (long long)LSEQ * LSEQ,
        v_pts, 576LL, 48LL,
        o_geom, 576LL, 48LL,
        LSEQ, LSEQ);

    // o_pair[i,h,c] = sum_j att[h,i,j] z[i,j,c] (batched over i, M=12 padded)
    wmma_gemm_batched<<<dim3(CZD/16, 1, LSEQ), wv, 0, stream>>>(
        att, (long long)LSEQ * LSEQ, (long long)LSEQ,
        z, (long long)CZD, (long long)LSEQ * CZD,
        concat + 864, (long long)CZD, 2400LL,
        NH, LSEQ);

    // invert frames into concat[:, 192:864]
    frame_inv_kernel<<<(LSEQ * NH * PVN + 255) / 256, 256, 0, stream>>>(
        o_geom, rot, trans, concat);

    // final: concat(320x2400) @ Wo(2400x384) + bo
    wmma_gemm_bias<<<dim3(CSD/16, mt), wv, 0, stream>>>(
        concat, Wo, bo, out, 2400, 2400, CSD, CSD);
}